// SO3Reparameterize_9826885174144
// MI455X (gfx1250) — compile-verified
//
#include <hip/hip_runtime.h>
#include <math.h>

typedef __attribute__((ext_vector_type(2))) float v2f;
typedef __attribute__((ext_vector_type(8))) float v8f;

#define DIN 1024
#define NPAIR (DIN / 2)          // 512 K-pairs
#define BSTRIDE 10               // 9 real cols + 1 dummy slot; keeps halves bank-disjoint
#define TILE_M 16
#define WAVES_PER_BLOCK 8
#define BLOCK_THREADS (WAVES_PER_BLOCK * 32)

__device__ __forceinline__ float softplus_f(float x) {
    // matches jax.nn.softplus = log1p(exp(x)) with overflow guard
    return x > 20.0f ? x : log1pf(expf(x));
}

__device__ __forceinline__ void rodrigues(float vx, float vy, float vz, float* R) {
    const float th  = sqrtf(vx * vx + vy * vy + vz * vz);
    const float inv = 1.0f / th;                 // reference has no eps guard either
    const float ux = vx * inv, uy = vy * inv, uz = vz * inv;
    const float s = sinf(th);
    const float c = cosf(th);
    const float t = 1.0f - c;
    R[0] = c + t * ux * ux;      R[1] = t * ux * uy - s * uz; R[2] = t * ux * uz + s * uy;
    R[3] = t * uy * ux + s * uz; R[4] = c + t * uy * uy;      R[5] = t * uy * uz - s * ux;
    R[6] = t * uz * ux - s * uy; R[7] = t * uz * uy + s * ux; R[8] = c + t * uz * uz;
}

__device__ __forceinline__ float pick_w(const float* __restrict__ Wmu,
                                        const float* __restrict__ Wd,
                                        const float* __restrict__ Wl,
                                        int k, int c) {
    if (c < 3) return Wmu[k * 3 + c];
    if (c < 6) return Wd[k * 3 + (c - 3)];
    return Wl[k * 3 + (c - 6)];
}

__global__ __launch_bounds__(BLOCK_THREADS)
void so3_fused_kernel(const float* __restrict__ x,   const float* __restrict__ eps,
                      const float* __restrict__ Wmu, const float* __restrict__ bmu,
                      const float* __restrict__ Wd,  const float* __restrict__ bd,
                      const float* __restrict__ Wl,  const float* __restrict__ bl,
                      float* __restrict__ out, int B, int n)
{
    // Packed K-pair weights: lw2[p*BSTRIDE + c] = { W[2p][c], W[2p+1][c] }  (40 KB)
    __shared__ v2f lw2[NPAIR * BSTRIDE];
    // Per-wave 16x16 accumulator staging tile (8 KB)
    __shared__ float tileBuf[WAVES_PER_BLOCK * TILE_M * TILE_M];

    const int tid = threadIdx.x;
    for (int i = tid; i < NPAIR * BSTRIDE; i += BLOCK_THREADS) {
        const int p = i / BSTRIDE;
        const int c = i - p * BSTRIDE;
        v2f w;
        if (c < 9) {
            w.x = pick_w(Wmu, Wd, Wl, 2 * p,     c);
            w.y = pick_w(Wmu, Wd, Wl, 2 * p + 1, c);
        } else {
            w.x = 0.0f; w.y = 0.0f;   // dummy slot (D cols 9..15 are never read)
        }
        lw2[i] = w;
    }
    __syncthreads();

    const int lane   = tid & 31;
    const int wave   = tid >> 5;
    const int nTiles = (B + TILE_M - 1) / TILE_M;
    int tile = blockIdx.x * WAVES_PER_BLOCK + wave;
    if (tile >= nTiles) tile = nTiles - 1;     // uniform per wave: duplicate work, no divergence
    const int row0 = tile * TILE_M;

    // A-operand (16x4 f32): lanes 0-15 hold M=lane, K = k+{0,1};
    //                       lanes 16-31 hold M=lane-16, K = k+{2,3}
    int aRow = row0 + (lane & 15);
    if (aRow >= B) aRow = B - 1;
    const int   half = lane >> 4;              // 0 or 1
    const int   kOff = half << 1;
    const float* xA  = x + (size_t)aRow * DIN + kOff;

    // B-operand (4x16 f32), mirrored K split. Column clamped into the packed array;
    // columns 9..15 read the dummy slot (their D columns are discarded).
    const int bColRaw = lane & 15;
    const int bCol    = bColRaw < 9 ? bColRaw : 9;
    const v2f* bp     = lw2 + half * BSTRIDE + bCol;   // + p*BSTRIDE per K-pair step

    v8f acc = {};
#pragma unroll 8
    for (int k = 0; k < DIN; k += 4) {
        v2f a = *(const v2f*)(xA + k);                 // global_load_b64
        v2f b = bp[(k >> 1) * BSTRIDE];                // ds_load_b64, unconditional
        acc = __builtin_amdgcn_wmma_f32_16x16x4_f32(
            /*neg_a=*/false, a, /*neg_b=*/false, b,
            /*c_mod=*/(short)0, acc, /*reuse_a=*/false, /*reuse_b=*/false);
    }

    // C/D layout: VGPR j: lanes 0-15 -> (M=j, N=lane), lanes 16-31 -> (M=j+8, N=lane-16)
    float* T = tileBuf + wave * (TILE_M * TILE_M);
    {
        const int rbase = half * 8;
        const int colT  = bColRaw;
#pragma unroll
        for (int j = 0; j < 8; ++j)
            T[(rbase + j) * TILE_M + colT] = acc[j];
    }
    // Intra-wave LDS visibility via gfx1250 split DS counter
    asm volatile("s_wait_dscnt 0" ::: "memory");

    if (lane < TILE_M) {
        const int row = row0 + lane;
        if (row < B) {
            const float* r = T + lane * TILE_M;

            const float m0 = r[0] + bmu[0], m1 = r[1] + bmu[1], m2 = r[2] + bmu[2];
            const float sd0 = sqrtf(softplus_f(r[3] + bd[0]));
            const float sd1 = sqrtf(softplus_f(r[4] + bd[1]));
            const float sd2 = sqrtf(softplus_f(r[5] + bd[2]));
            const float l0 = r[6] + bl[0], l1 = r[7] + bl[1], l2 = r[8] + bl[2];

            float Rm[9];
            rodrigues(m0, m1, m2, Rm);

            for (int j = 0; j < n; ++j) {
                const float* e = eps + ((size_t)j * B + row) * 3;
                const float s0 = sd0 * e[0];
                const float s1 = sd1 * e[1];
                const float s2 = sd2 * e[2];
                // v = L @ s, unit lower-triangular L = [[1,0,0],[l0,1,0],[l1,l2,1]]
                const float v0 = s0;
                const float v1 = l0 * s0 + s1;
                const float v2 = l1 * s0 + l2 * s1 + s2;

                float Rv[9];
                rodrigues(v0, v1, v2, Rv);

                float* o = out + ((size_t)j * B + row) * 9;
#pragma unroll
                for (int i = 0; i < 3; ++i) {
#pragma unroll
                    for (int c = 0; c < 3; ++c) {
                        o[i * 3 + c] = Rm[i * 3 + 0] * Rv[0 * 3 + c]
                                     + Rm[i * 3 + 1] * Rv[1 * 3 + c]
                                     + Rm[i * 3 + 2] * Rv[2 * 3 + c];
                    }
                }
            }
        }
    }
}

extern "C" void kernel_launch(void* const* d_in, const int* in_sizes, int n_in,
                              void* d_out, int out_size, void* d_ws, size_t ws_size,
                              hipStream_t stream) {
    const float* x   = (const float*)d_in[0];
    const float* eps = (const float*)d_in[1];
    const float* Wmu = (const float*)d_in[2];
    const float* bmu = (const float*)d_in[3];
    const float* Wd  = (const float*)d_in[4];
    const float* bd  = (const float*)d_in[5];
    const float* Wl  = (const float*)d_in[6];
    const float* bl  = (const float*)d_in[7];
    float* out = (float*)d_out;

    const int B = in_sizes[0] / DIN;          // 65536
    const int n = in_sizes[1] / (B * 3);      // 1

    const int nTiles  = (B + TILE_M - 1) / TILE_M;
    const int nBlocks = (nTiles + WAVES_PER_BLOCK - 1) / WAVES_PER_BLOCK;

    so3_fused_kernel<<<nBlocks, BLOCK_THREADS, 0, stream>>>(
        x, eps, Wmu, bmu, Wd, bd, Wl, bl, out, B, n);
}